// GraphModule_59012850647687
// MI455X (gfx1250) — compile-verified
//
#include <hip/hip_runtime.h>
#include <hip/hip_bf16.h>

typedef float v2f __attribute__((ext_vector_type(2)));
typedef float v8f __attribute__((ext_vector_type(8)));

#define D_H 256

// -------------------------------------------------------------------------
// aggr[i] = (1 + eps[layer]) * h[i]   (self term of GIN, element-wise)
// -------------------------------------------------------------------------
__global__ void init_aggr(const float* __restrict__ h, float* __restrict__ aggr,
                          const float* __restrict__ eps_all, int layer, int total) {
    int i = blockIdx.x * blockDim.x + threadIdx.x;
    float s = 1.0f + eps_all[layer];
    if (i < total) aggr[i] = s * h[i];
}

// -------------------------------------------------------------------------
// aggr[dst[e], :] += h[src[e], :]   — 32 lanes per edge, lane-strided so
// both the gather loads and the float atomics are coalesced per wave.
// -------------------------------------------------------------------------
__global__ void scatter_edges(const float* __restrict__ h, float* __restrict__ aggr,
                              const int* __restrict__ ei, int E, int d) {
    int tid  = blockIdx.x * blockDim.x + threadIdx.x;
    int edge = tid >> 5;
    int lane = tid & 31;
    if (edge >= E) return;
    int s = ei[edge];        // edge_index[0][e] : source (message sender)
    int t = ei[E + edge];    // edge_index[1][e] : destination (aggregator)
    const float* xs = h + (size_t)s * d;
    float*       ad = aggr + (size_t)t * d;
#pragma unroll
    for (int c = lane; c < d; c += 32) {
        unsafeAtomicAdd(&ad[c], xs[c]);   // -> global_atomic_add_f32
    }
}

// -------------------------------------------------------------------------
// out[N, 256] = relu( H[N, K] @ W[256, K]^T + bias )
// One wave computes a 16(row) x 64(col) strip via 4 accumulator tiles of
// v_wmma_f32_16x16x4_f32; 4 waves/block cover 16 x 256.
//
// Fragment layouts (32-bit, wave32):
//   A 16x4 : lane = m + 16*(k>>1), vgpr = k&1  -> lane holds H[m][k0+2*khi + {0,1}]
//   B 4x16 : lane = n + 16*(k>>1), vgpr = k&1  -> lane holds W[n][k0+2*khi + {0,1}]
//   C 16x16: lane = n + 16*half,   vgpr r -> M = r + 8*half
// -------------------------------------------------------------------------
__global__ void __launch_bounds__(128)
gin_gemm(const float* __restrict__ H, const float* __restrict__ W,
         const float* __restrict__ bias, float* __restrict__ out, int K) {
    const int lane = threadIdx.x & 31;
    const int wave = threadIdx.x >> 5;
    const int row0 = blockIdx.x * 16;
    const int colbase = wave * 64;

    const int mrow = lane & 15;   // A row within tile / B col within tile
    const int khi  = lane >> 4;   // high half of K quad

    const float* aptr = H + (size_t)(row0 + mrow) * K + 2 * khi;
    const float* bptr0 = W + (size_t)(colbase +  0 + mrow) * K + 2 * khi;
    const float* bptr1 = W + (size_t)(colbase + 16 + mrow) * K + 2 * khi;
    const float* bptr2 = W + (size_t)(colbase + 32 + mrow) * K + 2 * khi;
    const float* bptr3 = W + (size_t)(colbase + 48 + mrow) * K + 2 * khi;

    v8f c0 = {}, c1 = {}, c2 = {}, c3 = {};

#pragma unroll 4
    for (int k0 = 0; k0 < K; k0 += 4) {
        v2f a  = *(const v2f*)(aptr  + k0);
        v2f b0 = *(const v2f*)(bptr0 + k0);
        v2f b1 = *(const v2f*)(bptr1 + k0);
        v2f b2 = *(const v2f*)(bptr2 + k0);
        v2f b3 = *(const v2f*)(bptr3 + k0);
        c0 = __builtin_amdgcn_wmma_f32_16x16x4_f32(false, a, false, b0, (short)0, c0, false, false);
        c1 = __builtin_amdgcn_wmma_f32_16x16x4_f32(false, a, false, b1, (short)0, c1, false, false);
        c2 = __builtin_amdgcn_wmma_f32_16x16x4_f32(false, a, false, b2, (short)0, c2, false, false);
        c3 = __builtin_amdgcn_wmma_f32_16x16x4_f32(false, a, false, b3, (short)0, c3, false, false);
    }

    // Epilogue: bias + ReLU, store per C/D layout.
    const int ncol = lane & 15;
    const int half = lane >> 4;   // M offset +8 for lanes 16..31
    v8f acc[4] = {c0, c1, c2, c3};
#pragma unroll
    for (int t = 0; t < 4; ++t) {
        int col = colbase + t * 16 + ncol;
        float bv = bias[col];
#pragma unroll
        for (int r = 0; r < 8; ++r) {
            int row = row0 + r + 8 * half;
            float v = acc[t][r] + bv;
            out[(size_t)row * D_H + col] = v > 0.0f ? v : 0.0f;
        }
    }
}

// -------------------------------------------------------------------------
// Orchestration: ping-pong  d_out (current h)  <->  ws B1 (aggr) / B2 (h1)
// -------------------------------------------------------------------------
extern "C" void kernel_launch(void* const* d_in, const int* in_sizes, int n_in,
                              void* d_out, int out_size, void* d_ws, size_t ws_size,
                              hipStream_t stream) {
    const float* x       = (const float*)d_in[0];
    const int*   ei      = (const int*)  d_in[1];
    const float* eps_all = (const float*)d_in[2];
    const float* w0_l0   = (const float*)d_in[3];
    const float* b0_l0   = (const float*)d_in[4];
    const float* w1_l0   = (const float*)d_in[5];
    const float* b1_l0   = (const float*)d_in[6];
    const float* w0_rest = (const float*)d_in[7];   // [4, 256, 256]
    const float* b0_rest = (const float*)d_in[8];   // [4, 256]
    const float* w1_rest = (const float*)d_in[9];
    const float* b1_rest = (const float*)d_in[10];

    const int N = in_sizes[0] / 128;   // 50000
    const int E = in_sizes[1] / 2;     // 800000
    float* hout = (float*)d_out;

    float* B1 = (float*)d_ws;                       // aggr / h_pre  [N, 256]
    float* B2 = (float*)d_ws + (size_t)N * D_H;     // h1            [N, 256]

    const int gemmBlocks = N / 16;                  // 3125
    const int edgeBlocks = (E * 32 + 255) / 256;

    // ---- Layer 0 (input dim 128) ----
    {
        int tot = N * 128;
        init_aggr<<<(tot + 255) / 256, 256, 0, stream>>>(x, B1, eps_all, 0, tot);
        scatter_edges<<<edgeBlocks, 256, 0, stream>>>(x, B1, ei, E, 128);
        gin_gemm<<<gemmBlocks, 128, 0, stream>>>(B1, w0_l0, b0_l0, B2, 128);
        gin_gemm<<<gemmBlocks, 128, 0, stream>>>(B2, w1_l0, b1_l0, hout, 256);
    }

    // ---- Layers 1..4 (dim 256) ----
    for (int l = 1; l <= 4; ++l) {
        const float* w0 = w0_rest + (size_t)(l - 1) * D_H * D_H;
        const float* b0 = b0_rest + (size_t)(l - 1) * D_H;
        const float* w1 = w1_rest + (size_t)(l - 1) * D_H * D_H;
        const float* b1 = b1_rest + (size_t)(l - 1) * D_H;
        int tot = N * D_H;
        init_aggr<<<(tot + 255) / 256, 256, 0, stream>>>(hout, B1, eps_all, l, tot);
        scatter_edges<<<edgeBlocks, 256, 0, stream>>>(hout, B1, ei, E, 256);
        gin_gemm<<<gemmBlocks, 128, 0, stream>>>(B1, w0, b0, B2, 256);
        gin_gemm<<<gemmBlocks, 128, 0, stream>>>(B2, w1, b1, hout, 256);
    }
}